// Transducer_28913719837048
// MI455X (gfx1250) — compile-verified
//
#include <hip/hip_runtime.h>
#include <hip/hip_bf16.h>

#define GLOBAL_AS __attribute__((address_space(1)))
#define LDS_AS    __attribute__((address_space(3)))

typedef int v4i __attribute__((vector_size(16)));  // matches builtin's V4i param

#if defined(__has_builtin)
#  if __has_builtin(__builtin_amdgcn_global_load_async_to_lds_b128)
#    define HAVE_ASYNC_LDS 1
#  endif
#endif
#ifndef HAVE_ASYNC_LDS
#  define HAVE_ASYNC_LDS 0
#endif

namespace {
constexpr int   kN = 8, kT = 200, kU = 50, kV = 500;
constexpr int   kRows  = kN * kT * kU;          // 80000 rows per tensor
constexpr int   kTasks = 2 * kRows;             // student + teacher
constexpr float kEps   = 1e-10f;
constexpr int   kWavesPerBlock = 8;             // 256 threads, wave32
constexpr int   kBlocks = 2000;
constexpr int   kNW    = kBlocks * kWavesPerBlock;  // 16000 waves
constexpr int   kIters = kTasks / kNW;              // exactly 10 rows per wave
static_assert(kTasks % kNW == 0, "exact division required for pipeline epilogue");
constexpr int   kDepth = 3;                     // triple-buffered row staging
}

template <int Cnt>
__device__ __forceinline__ void wait_asynccnt() {
#if defined(__has_builtin) && __has_builtin(__builtin_amdgcn_s_wait_asynccnt)
  __builtin_amdgcn_s_wait_asynccnt(Cnt);
#else
  asm volatile("s_wait_asynccnt %0" ::"i"(Cnt) : "memory");
#endif
}

// Stage one full 2000-byte row into LDS with 4 async b128 transfers
// (offsets 0/512/1024 all lanes; offset 1536 with lanes 0..28 covering the
// 464-byte tail). each lane moves 16 bytes global->LDS per transfer,
// tracked with ASYNCcnt (4 increments per row, completes in order).
__device__ __forceinline__ void stage_row(const float* gsrc, float* ldsb, int lane) {
#if HAVE_ASYNC_LDS
  GLOBAL_AS v4i* g = (GLOBAL_AS v4i*)(gsrc + lane * 4);
  LDS_AS    v4i* l = (LDS_AS    v4i*)(ldsb + lane * 4);
  __builtin_amdgcn_global_load_async_to_lds_b128(g, l, 0,    0);
  __builtin_amdgcn_global_load_async_to_lds_b128(g, l, 512,  0);
  __builtin_amdgcn_global_load_async_to_lds_b128(g, l, 1024, 0);
  if (lane < 29)
    __builtin_amdgcn_global_load_async_to_lds_b128(g, l, 1536, 0);
#else
  const float4* g4 = (const float4*)(gsrc);
  float4*       l4 = (float4*)(ldsb);
  l4[lane]      = g4[lane];
  l4[lane + 32] = g4[lane + 32];
  l4[lane + 64] = g4[lane + 64];
  if (lane < 29) l4[lane + 96] = g4[lane + 96];
#endif
}

__global__ __launch_bounds__(256, 1) void transducer_kd_kernel(
    const float* __restrict__ logits,
    const float* __restrict__ teacher,
    const int*   __restrict__ y,
    const int*   __restrict__ xlen,
    const int*   __restrict__ ylen,
    float*       __restrict__ out) {
  __shared__ __align__(16) float stage[kWavesPerBlock][kDepth][512];  // 48 KB

  const int lane  = threadIdx.x & 31;
  const int wave  = threadIdx.x >> 5;
  const int gwave = blockIdx.x * kWavesPerBlock + wave;

  auto rowptr = [&](int t) -> const float* {
    const int r = (t >= kRows) ? t - kRows : t;
    return ((t >= kRows) ? teacher : logits) + (size_t)r * kV;
  };

  // prologue: prefetch rows 0 and 1 of this wave's task list
  stage_row(rowptr(gwave),       &stage[wave][0][0], lane);
  stage_row(rowptr(gwave + kNW), &stage[wave][1][0], lane);

#pragma unroll 1
  for (int it = 0; it < kIters; ++it) {
    const int  task      = gwave + it * kNW;
    const bool isTeacher = task >= kRows;
    const int  row       = isTeacher ? task - kRows : task;

    // keep prefetch distance 2; exact ASYNCcnt bookkeeping (4 ops per row)
    if (it < kIters - 2) {
      stage_row(rowptr(task + 2 * kNW), &stage[wave][(it + 2) % kDepth][0], lane);
      wait_asynccnt<8>();   // rows it+1, it+2 may be in flight; row it is done
    } else if (it == kIters - 2) {
      wait_asynccnt<4>();   // only row it+1 still in flight
    } else {
      wait_asynccnt<0>();
    }

    const float*  lbuf = &stage[wave][it % kDepth][0];
    const float4* l4   = (const float4*)lbuf;

    // 16 values per lane: v = lane*4 + {0,128,256,384} (+0..3)
    float4 a0 = l4[lane];
    float4 a1 = l4[lane + 32];
    float4 a2 = l4[lane + 64];
    const bool tv = lane < 29;  // floats 384..499 only
    float4 a3 = make_float4(-3.0e38f, -3.0e38f, -3.0e38f, -3.0e38f);
    if (tv) a3 = l4[lane + 96];

    // ---- row max (wave32 xor-shuffle reduction) ----
    float mx = fmaxf(fmaxf(fmaxf(a0.x, a0.y), fmaxf(a0.z, a0.w)),
                     fmaxf(fmaxf(fmaxf(a1.x, a1.y), fmaxf(a1.z, a1.w)),
                           fmaxf(fmaxf(a2.x, a2.y), fmaxf(a2.z, a2.w))));
    mx = fmaxf(mx, fmaxf(fmaxf(a3.x, a3.y), fmaxf(a3.z, a3.w)));
#pragma unroll
    for (int off = 16; off >= 1; off >>= 1)
      mx = fmaxf(mx, __shfl_xor(mx, off, 32));

    // ---- sum of exp ----
    float ssum = __expf(a0.x - mx) + __expf(a0.y - mx) +
                 __expf(a0.z - mx) + __expf(a0.w - mx) +
                 __expf(a1.x - mx) + __expf(a1.y - mx) +
                 __expf(a1.z - mx) + __expf(a1.w - mx) +
                 __expf(a2.x - mx) + __expf(a2.y - mx) +
                 __expf(a2.z - mx) + __expf(a2.w - mx);
    if (tv)
      ssum += __expf(a3.x - mx) + __expf(a3.y - mx) +
              __expf(a3.z - mx) + __expf(a3.w - mx);
#pragma unroll
    for (int off = 16; off >= 1; off >>= 1)
      ssum += __shfl_xor(ssum, off, 32);

    if (lane == 0) {
      const int nn = row / (kT * kU);
      const int rr = row % (kT * kU);
      const int tt = rr / kU;
      const int uu = rr % kU;
      const int yv = y[nn * kU + uu];  // in [1, 500)

      const float ly  = lbuf[yv];  // full row is in LDS: direct gather
      const float l0  = lbuf[0];
      const float inv = 1.0f / ssum;
      const float py  = __expf(ly - mx) * inv;
      const float pb  = __expf(l0 - mx) * inv;
      const float pr  = 1.0f - py - pb;
      float o0, o1, o2;
      if (!isTeacher) {
        o0 = __logf(fminf(fmaxf(py, kEps), 1.0f));
        o1 = __logf(fminf(fmaxf(pb, kEps), 1.0f));
        o2 = __logf(fminf(fmaxf(pr, kEps), 1.0f));
      } else {
        o0 = py;
        o1 = pb;
        o2 = (pr < 0.0f) ? kEps : pr;
      }
      const float m = ((tt < xlen[nn]) && (uu < ylen[nn])) ? 1.0f : 0.0f;
      const size_t o =
          ((((size_t)(isTeacher ? 1 : 0) * kN + nn) * kT + tt) * kU + uu) * 3;
      out[o + 0] = o0 * m;
      out[o + 1] = o1 * m;
      out[o + 2] = o2 * m;
    }
  }
}

extern "C" void kernel_launch(void* const* d_in, const int* in_sizes, int n_in,
                              void* d_out, int out_size, void* d_ws, size_t ws_size,
                              hipStream_t stream) {
  (void)in_sizes; (void)n_in; (void)out_size; (void)d_ws; (void)ws_size;
  const float* logits  = (const float*)d_in[0];
  const float* teacher = (const float*)d_in[1];
  const int*   yy      = (const int*)d_in[2];
  const int*   xlen    = (const int*)d_in[3];
  const int*   ylen    = (const int*)d_in[4];
  float*       out     = (float*)d_out;

  hipLaunchKernelGGL(transducer_kd_kernel, dim3(kBlocks), dim3(256), 0, stream,
                     logits, teacher, yy, xlen, ylen, out);
}